// MessagePassingLayer_56221121904751
// MI455X (gfx1250) — compile-verified
//
#include <hip/hip_runtime.h>
#include <hip/hip_bf16.h>

// Problem constants (B, N, F_IN, F_OUT) = (4, 512, 128, 128)
#define BN    2048      // B*N rows
#define NN    512       // neighbors per row
#define FF    128       // feature dim (F_IN == F_OUT)

typedef __attribute__((ext_vector_type(2))) float v2f;
typedef __attribute__((ext_vector_type(8))) float v8f;

// ---------------------------------------------------------------------------
// Kernel A: xi = x @ W_msg[0:128,:], xj = x @ W_msg[128:256,:]
// One 16x16 f32 output tile per wave via V_WMMA_F32_16X16X4_F32 (K-loop of 32).
// Grid: 2048 tiles (128 M-tiles x 16 N-tiles) / 8 waves per block = 256 blocks.
// ---------------------------------------------------------------------------
__global__ void gemm_msg_kernel(const float* __restrict__ x,
                                const float* __restrict__ W_msg,
                                float* __restrict__ xi,
                                float* __restrict__ xj) {
    const int wid  = threadIdx.x >> 5;
    const int lane = threadIdx.x & 31;
    const int gt   = blockIdx.x * 8 + wid;       // global tile id, 0..2047
    const int m0   = (gt >> 4) * 16;             // row tile base
    const int nt   = gt & 15;                    // 0..7 -> xi, 8..15 -> xj

    const float* Bbase = (nt < 8) ? W_msg : (W_msg + FF * FF);
    float*       Obase = (nt < 8) ? xi : xj;
    const int    o0    = (nt & 7) * 16;

    const int hi = lane >> 4;                    // 0: K,K+1  1: K+2,K+3
    const int ml = lane & 15;

    const float* pa = x + (size_t)(m0 + ml) * FF + 2 * hi;      // A: row-major
    const float* pb = Bbase + (size_t)(2 * hi) * FF + (o0 + ml); // B: row-major KxN

    v8f c = {};
#pragma unroll 4
    for (int k = 0; k < FF; k += 4) {
        v2f a = *(const v2f*)(pa + k);
        v2f b = { pb[(size_t)k * FF], pb[(size_t)k * FF + FF] };
        c = __builtin_amdgcn_wmma_f32_16x16x4_f32(
                /*neg_a=*/false, a, /*neg_b=*/false, b,
                /*c_mod=*/(short)0, c, /*reuse_a=*/false, /*reuse_b=*/false);
    }

#pragma unroll
    for (int r = 0; r < 8; ++r)
        Obase[(size_t)(m0 + r + 8 * hi) * FF + o0 + ml] = c[r];
}

// ---------------------------------------------------------------------------
// Kernel B: agg[b,i,o] = sum_j adj[b,i,j] * relu(xi[b,i,o] + xj[b,j,o] + b_msg[o])
// One block per (b,i); 128 threads = one output feature each.
// adj row staged into LDS; per-j scalar is a broadcast read, branch is uniform
// across the wave so the 90%-zero adjacency skips both the load and the math.
// ---------------------------------------------------------------------------
__global__ void aggregate_kernel(const float* __restrict__ xi,
                                 const float* __restrict__ xj,
                                 const float* __restrict__ adj,
                                 const float* __restrict__ b_msg,
                                 float* __restrict__ agg) {
    __shared__ float sadj[NN];
    const int bi = blockIdx.x;            // b*512 + i
    const int o  = threadIdx.x;           // 0..127
    const int b  = bi >> 9;

    const float* arow = adj + (size_t)bi * NN;
#pragma unroll
    for (int t = 0; t < NN / 128; ++t)
        sadj[o + 128 * t] = arow[o + 128 * t];
    __syncthreads();

    const float  base = xi[(size_t)bi * FF + o] + b_msg[o];
    const float* pxj  = xj + ((size_t)(b << 9)) * FF + o;

    float acc = 0.f;
    for (int j = 0; j < NN; ++j) {
        float a = sadj[j];
        if (a != 0.0f)
            acc = fmaf(a, fmaxf(base + pxj[(size_t)j * FF], 0.f), acc);
    }
    agg[(size_t)bi * FF + o] = acc;
}

// ---------------------------------------------------------------------------
// Kernel C: out = relu(concat(x, agg) @ W_upd + b_upd)
// Same WMMA tiling; K=256 split into two 128-deep phases (x then agg).
// Bias + ReLU fused into the store epilogue.
// Grid: 1024 tiles (128 x 8) / 8 waves per block = 128 blocks.
// ---------------------------------------------------------------------------
__global__ void gemm_upd_kernel(const float* __restrict__ x,
                                const float* __restrict__ agg,
                                const float* __restrict__ W_upd,
                                const float* __restrict__ b_upd,
                                float* __restrict__ out) {
    const int wid  = threadIdx.x >> 5;
    const int lane = threadIdx.x & 31;
    const int gt   = blockIdx.x * 8 + wid;       // 0..1023
    const int m0   = (gt >> 3) * 16;
    const int o0   = (gt & 7) * 16;

    const int hi = lane >> 4;
    const int ml = lane & 15;

    const float* pa0 = x   + (size_t)(m0 + ml) * FF + 2 * hi;
    const float* pa1 = agg + (size_t)(m0 + ml) * FF + 2 * hi;
    const float* pb  = W_upd + (size_t)(2 * hi) * FF + (o0 + ml); // (256,128) row-major

    v8f c = {};
#pragma unroll 4
    for (int k = 0; k < FF; k += 4) {            // phase 1: A = x, W rows 0..127
        v2f a = *(const v2f*)(pa0 + k);
        v2f b = { pb[(size_t)k * FF], pb[(size_t)k * FF + FF] };
        c = __builtin_amdgcn_wmma_f32_16x16x4_f32(false, a, false, b,
                                                  (short)0, c, false, false);
    }
#pragma unroll 4
    for (int k = 0; k < FF; k += 4) {            // phase 2: A = agg, W rows 128..255
        v2f a = *(const v2f*)(pa1 + k);
        v2f b = { pb[(size_t)(k + FF) * FF], pb[(size_t)(k + FF) * FF + FF] };
        c = __builtin_amdgcn_wmma_f32_16x16x4_f32(false, a, false, b,
                                                  (short)0, c, false, false);
    }

    const float bias = b_upd[o0 + ml];
#pragma unroll
    for (int r = 0; r < 8; ++r)
        out[(size_t)(m0 + r + 8 * hi) * FF + o0 + ml] = fmaxf(c[r] + bias, 0.f);
}

// ---------------------------------------------------------------------------
extern "C" void kernel_launch(void* const* d_in, const int* in_sizes, int n_in,
                              void* d_out, int out_size, void* d_ws, size_t ws_size,
                              hipStream_t stream) {
    const float* x     = (const float*)d_in[0];
    const float* adj   = (const float*)d_in[1];
    const float* W_msg = (const float*)d_in[2];
    const float* b_msg = (const float*)d_in[3];
    const float* W_upd = (const float*)d_in[4];
    const float* b_upd = (const float*)d_in[5];
    float* out = (float*)d_out;

    // Workspace: xi (1MB) | xj (1MB) | agg (1MB)  -> 3MB total
    float* xi  = (float*)d_ws;
    float* xj  = xi + (size_t)BN * FF;
    float* agg = xj + (size_t)BN * FF;

    // A: 2048 wave-tiles, 8 waves (256 threads) per block
    gemm_msg_kernel<<<256, 256, 0, stream>>>(x, W_msg, xi, xj);
    // B: one block per (b,i) row
    aggregate_kernel<<<BN, 128, 0, stream>>>(xi, xj, adj, b_msg, agg);
    // C: 1024 wave-tiles
    gemm_upd_kernel<<<128, 256, 0, stream>>>(x, agg, W_upd, b_upd, out);
}